// NodeNetwork_10823317585951
// MI455X (gfx1250) — compile-verified
//
#include <hip/hip_runtime.h>

#define N_NODES 50000
#define N_EDGES 1000000
#define D 64
#define EPS 1e-5f

typedef float v2f __attribute__((ext_vector_type(2)));
typedef float v8f __attribute__((ext_vector_type(8)));

// ---------------------------------------------------------------------------
// Clear the mi/mo accumulators (must re-zero every call: graph replay).
// ---------------------------------------------------------------------------
__global__ void zero_kernel(float* __restrict__ p, int n) {
    int i = blockIdx.x * blockDim.x + threadIdx.x;
    int stride = gridDim.x * blockDim.x;
    for (; i < n; i += stride) p[i] = 0.0f;
}

// ---------------------------------------------------------------------------
// Edge scatter: mi[end] += e*x[start];  mo[start] += e*x[end]
// 16 lanes cooperate on one edge; each lane handles 4 contiguous features
// (float4 load = 256B coalesced per edge row, 8 global_atomic_add_f32).
// Accumulators + x fit in L2 (192MB), so atomics resolve at L2 atomic units.
// ---------------------------------------------------------------------------
__global__ void edge_scatter(const float* __restrict__ x,
                             const float* __restrict__ e,
                             const long long* __restrict__ ei,  // (2, E) int64
                             float* __restrict__ mi,
                             float* __restrict__ mo) {
    int w = blockIdx.x * blockDim.x + threadIdx.x;
    const int total = N_EDGES * 16;
    const int stride = gridDim.x * blockDim.x;
    for (; w < total; w += stride) {
        const int edge = w >> 4;
        const int c4   = (w & 15) * 4;
        const int s = (int)ei[edge];            // edge_index[0][edge]
        const int t = (int)ei[N_EDGES + edge];  // edge_index[1][edge]
        const float we = e[edge];
        const float4 xs = *(const float4*)(x + s * D + c4);
        const float4 xt = *(const float4*)(x + t * D + c4);
        float* miRow = mi + t * D + c4;
        float* moRow = mo + s * D + c4;
        atomicAdd(miRow + 0, we * xs.x);
        atomicAdd(miRow + 1, we * xs.y);
        atomicAdd(miRow + 2, we * xs.z);
        atomicAdd(miRow + 3, we * xs.w);
        atomicAdd(moRow + 0, we * xt.x);
        atomicAdd(moRow + 1, we * xt.y);
        atomicAdd(moRow + 2, we * xt.z);
        atomicAdd(moRow + 3, we * xt.w);
    }
}

// ---------------------------------------------------------------------------
// Fused GEMM (via V_WMMA_F32_16X16X4_F32) + bias + LayerNorm + ReLU.
// Block = 128 threads = 4 wave32; each wave computes one 16x16 output tile,
// the 4 waves cover the 64 output columns of a 16-row stripe.
// nseg==3 reads the virtual concat [s0|s1|s2] (layer 1), nseg==1 reads s0.
// A-frag (16x4 f32): lane<16 -> K={k,k+1}, lane>=16 -> K={k+2,k+3}, M=lane&15.
// B-frag (4x16 f32): VGPR v holds rows K={v, v+2}, N=lane&15.
// C (16x16 f32): VGPR i holds M=i (lanes 0-15) / M=i+8 (lanes 16-31), N=lane&15.
// ---------------------------------------------------------------------------
__global__ __launch_bounds__(128)
void gemm_ln_relu(const float* __restrict__ s0,
                  const float* __restrict__ s1,
                  const float* __restrict__ s2,
                  int nseg,
                  const float* __restrict__ W,      // (nseg*64, 64) row-major
                  const float* __restrict__ bias,   // (64,)
                  const float* __restrict__ gamma,  // (64,)
                  const float* __restrict__ beta,   // (64,)
                  float* __restrict__ out) {        // (N, 64)
    __shared__ float tile[16][D + 1];
    __shared__ float smu[16];
    __shared__ float srs[16];

    const int tid   = threadIdx.x;
    const int lane  = tid & 31;
    const int wave  = tid >> 5;
    const int m     = lane & 15;      // row (A) / col (B) within tile
    const int khalf = lane >> 4;      // 0: K pair {0,1}; 1: K pair {2,3}
    const int rowBase = blockIdx.x * 16;
    const int colBase = wave * 16;

    v8f acc = {};
    for (int seg = 0; seg < nseg; ++seg) {
        const float* src  = (seg == 0) ? s0 : (seg == 1) ? s1 : s2;
        const float* arow = src + (rowBase + m) * D;
        const float* wseg = W + seg * D * D;
        for (int kk = 0; kk < D; kk += 4) {
            v2f a, b;
            a.x = arow[kk + 2 * khalf + 0];
            a.y = arow[kk + 2 * khalf + 1];
            b.x = wseg[(kk + 2 * khalf + 0) * D + colBase + m];
            b.y = wseg[(kk + 2 * khalf + 1) * D + colBase + m];
            // (neg_a, A, neg_b, B, c_mod, C, reuse_a, reuse_b)
            acc = __builtin_amdgcn_wmma_f32_16x16x4_f32(
                false, a, false, b, (short)0, acc, false, false);
        }
    }

    // Deposit tile (+bias) into LDS per the C/D VGPR layout.
#pragma unroll
    for (int i = 0; i < 8; ++i) {
        const int row = i + 8 * khalf;
        const int col = colBase + m;
        tile[row][col] = acc[i] + bias[col];
    }
    __syncthreads();

    // Per-row LayerNorm statistics.
    if (tid < 16) {
        float su = 0.0f, sq = 0.0f;
        for (int c = 0; c < D; ++c) {
            const float v = tile[tid][c];
            su += v;
            sq += v * v;
        }
        const float mu  = su * (1.0f / D);
        const float var = sq * (1.0f / D) - mu * mu;
        smu[tid] = mu;
        srs[tid] = rsqrtf(var + EPS);
    }
    __syncthreads();

    // Normalize + affine + ReLU, coalesced store. 16*64 = 1024 elems / 128 thr.
#pragma unroll
    for (int j = 0; j < 8; ++j) {
        const int eidx = tid + 128 * j;
        const int row  = eidx >> 6;
        const int col  = eidx & 63;
        float v = (tile[row][col] - smu[row]) * srs[row] * gamma[col] + beta[col];
        out[(rowBase + row) * D + col] = v > 0.0f ? v : 0.0f;
    }
}

// ---------------------------------------------------------------------------
// Launch
// ---------------------------------------------------------------------------
extern "C" void kernel_launch(void* const* d_in, const int* in_sizes, int n_in,
                              void* d_out, int out_size, void* d_ws, size_t ws_size,
                              hipStream_t stream) {
    (void)in_sizes; (void)n_in; (void)out_size; (void)ws_size;

    const float*     x  = (const float*)d_in[0];
    const float*     e  = (const float*)d_in[1];
    const long long* ei = (const long long*)d_in[2];  // int64 per reference
    const float* W1 = (const float*)d_in[3];
    const float* b1 = (const float*)d_in[4];
    const float* g1 = (const float*)d_in[5];
    const float* t1 = (const float*)d_in[6];
    const float* W2 = (const float*)d_in[7];
    const float* b2 = (const float*)d_in[8];
    const float* g2 = (const float*)d_in[9];
    const float* t2 = (const float*)d_in[10];
    const float* W3 = (const float*)d_in[11];
    const float* b3 = (const float*)d_in[12];
    const float* g3 = (const float*)d_in[13];
    const float* t3 = (const float*)d_in[14];
    const float* W4 = (const float*)d_in[15];
    const float* b4 = (const float*)d_in[16];
    const float* g4 = (const float*)d_in[17];
    const float* t4 = (const float*)d_in[18];

    float* ws = (float*)d_ws;
    float* mi = ws;                       // N*64
    float* mo = ws + (size_t)N_NODES * D; // N*64
    float* hA = ws + (size_t)2 * N_NODES * D;
    float* hB = ws + (size_t)3 * N_NODES * D;
    float* out = (float*)d_out;

    zero_kernel<<<2048, 256, 0, stream>>>(mi, 2 * N_NODES * D);
    edge_scatter<<<(N_EDGES * 16) / 256, 256, 0, stream>>>(x, e, ei, mi, mo);

    dim3 grid(N_NODES / 16);  // 3125, exact
    gemm_ln_relu<<<grid, 128, 0, stream>>>(mi, mo, x, 3, W1, b1, g1, t1, hA);
    gemm_ln_relu<<<grid, 128, 0, stream>>>(hA, hA, hA, 1, W2, b2, g2, t2, hB);
    gemm_ln_relu<<<grid, 128, 0, stream>>>(hB, hB, hB, 1, W3, b3, g3, t3, hA);
    gemm_ln_relu<<<grid, 128, 0, stream>>>(hA, hA, hA, 1, W4, b4, g4, t4, out);
}